// VectorQuantizerEMA_10127532884671
// MI455X (gfx1250) — compile-verified
//
#include <hip/hip_runtime.h>
#include <hip/hip_bf16.h>
#include <math.h>
#include <stdint.h>

// ---------------------------------------------------------------------------
// VQ-VAE (EMA) forward + codebook update for gfx1250 (MI455X), wave32.
//   inputs      [16,32,32,256] f32  -> flat [N=16384, D=256]
//   embeddings  [D=256, K=8192] f32
// Outputs (concat in d_out, f32): quantized (N*D) | loss (1) | perplexity (1)
//                                 | idx (N) | new_embeddings (D*K)
// ---------------------------------------------------------------------------

typedef __attribute__((ext_vector_type(2))) float v2f;
typedef __attribute__((ext_vector_type(8))) float v8f;

#define N_ROWS 16384
#define D_EMB  256
#define K_EMB  8192
#define DECAY  0.99f
#define ONE_M_DECAY 0.01f
#define EPS    1e-5f

// ---- workspace layout (float offsets) ----
#define WS_DW     0                          // dwT [K,D] floats (zeroed)
#define WS_CNT    (WS_DW + D_EMB * K_EMB)    // K floats         (zeroed)
#define WS_SCAL   (WS_CNT + K_EMB)           // 16 floats: [0]=loss [1]=n [2]=debias (zeroed)
#define WS_ENORM  (WS_SCAL + 16)             // K floats         (fully written)
#define WS_IDX    (WS_ENORM + K_EMB)         // N ints           (fully written)
#define WS_EMBT   (WS_IDX + N_ROWS)          // embT [K,D]       (fully written)
#define WS_ZERO_BYTES ((size_t)(WS_SCAL + 16) * 4)

// ---- gfx1250 async load-to-LDS path (per-lane LDS scatter, ASYNCcnt) ------
#if defined(__has_builtin)
#if __has_builtin(__builtin_amdgcn_global_load_async_to_lds_b32)
#define VQ_ASYNC_LDS 1
#endif
#endif
#ifndef VQ_ASYNC_LDS
#define VQ_ASYNC_LDS 0
#endif

__device__ __forceinline__ void vq_wait_async0() {
#if defined(__has_builtin) && __has_builtin(__builtin_amdgcn_s_wait_asynccnt)
  __builtin_amdgcn_s_wait_asynccnt(0);
#else
  asm volatile("s_wait_asynccnt 0x0" ::: "memory");
#endif
}

// ---------------------------------------------------------------------------
// 1) per-code squared norm  enorm[k] = sum_d emb[d,k]^2   (coalesced over k)
// ---------------------------------------------------------------------------
__global__ __launch_bounds__(256) void vq_enorm_kernel(
    const float* __restrict__ emb, float* __restrict__ enorm) {
  int k = blockIdx.x * 256 + threadIdx.x;
  float s = 0.f;
  for (int d = 0; d < D_EMB; ++d) {
    float v = emb[(size_t)d * K_EMB + k];
    s += v * v;
  }
  enorm[k] = s;
}

// ---------------------------------------------------------------------------
// 1b) tiled transpose embT[k,d] = emb[d,k]  (both sides coalesced via LDS)
// ---------------------------------------------------------------------------
__global__ __launch_bounds__(256) void vq_transpose_kernel(
    const float* __restrict__ emb, float* __restrict__ embT) {
  __shared__ float t[32][33];
  const int kb = blockIdx.x * 32;
  const int db = blockIdx.y * 32;
  const int tx = threadIdx.x & 31;
  const int ty = threadIdx.x >> 5;  // 0..7
#pragma unroll
  for (int r = 0; r < 32; r += 8)
    t[ty + r][tx] = emb[(size_t)(db + ty + r) * K_EMB + kb + tx];
  __syncthreads();
#pragma unroll
  for (int r = 0; r < 32; r += 8)
    embT[(size_t)(kb + ty + r) * D_EMB + db + tx] = t[tx][ty + r];
}

// ---------------------------------------------------------------------------
// 2) fused distance GEMM + argmin via V_WMMA_F32_16X16X4_F32.
//    Block = 256 threads = 8 waves; wave w owns rows m_base..m_base+15.
//    A (16x256 x-tile) lives in 128 VGPRs per lane. B (16 codes x 256 d)
//    is double-buffered in LDS (2 x 16KB), staged for tile kt+1 while tile
//    kt is consumed (async load-to-LDS when available).  B operands are
//    additionally double-buffered in registers in groups of 8 chunks so the
//    ds-load latency hides behind 8 WMMAs instead of stalling every pair.
//    LDS repack: element (d, k%16) -> [c=d/4][n=k%16][r=d%4]  (8B per lane)
// ---------------------------------------------------------------------------
#define NTILE (K_EMB / 16)

__device__ __forceinline__ void vq_stage_addrs(int i, int kt, int* gidx,
                                               int* lidx) {
  int d = i >> 4;
  int n = i & 15;
  *gidx = d * K_EMB + kt * 16 + n;          // global float index
  *lidx = (d >> 2) * 64 + n * 4 + (d & 3);  // lds float index
}

__global__ __launch_bounds__(256) void vq_argmin_kernel(
    const float* __restrict__ x,      // [N, D]
    const float* __restrict__ emb,    // [D, K]
    const float* __restrict__ enorm,  // [K]
    int*   __restrict__ idx_out,      // [N] (workspace)
    float* __restrict__ idx_f_out) {  // [N] (d_out section)
  __shared__ float lds_b[2][16 * D_EMB];

  const int tid  = threadIdx.x;
  const int wave = tid >> 5;
  const int lane = tid & 31;
  const int half = lane >> 4;  // 0: lanes 0-15, 1: lanes 16-31
  const int l16  = lane & 15;
  const int m_base = (blockIdx.x * 8 + wave) * 16;

  // Preload this wave's 16x256 A tile into registers (read once).
  v2f a[64];
  {
    const float* xr = x + (size_t)(m_base + l16) * D_EMB;
#pragma unroll
    for (int c = 0; c < 64; ++c) {
      int col = 4 * c + 2 * half;
      a[c].x = xr[col];
      a[c].y = xr[col + 1];
    }
  }

  float best[8];
  int   bidx[8];
#pragma unroll
  for (int j = 0; j < 8; ++j) { best[j] = 3.4e38f; bidx[j] = 0; }

  const int boff = l16 * 4 + 2 * half;

  // ---- prologue: stage tile 0 into buffer 0 ----
#if VQ_ASYNC_LDS
  {
#pragma unroll
    for (int t = 0; t < 16; ++t) {
      int gi, li;
      vq_stage_addrs(t * 256 + tid, 0, &gi, &li);
      __builtin_amdgcn_global_load_async_to_lds_b32(
          (int*)(emb + gi), (int*)&lds_b[0][li], 0, 0);
    }
  }
  vq_wait_async0();
#else
  {
#pragma unroll
    for (int t = 0; t < 16; ++t) {
      int gi, li;
      vq_stage_addrs(t * 256 + tid, 0, &gi, &li);
      lds_b[0][li] = emb[gi];
    }
  }
#endif
  __syncthreads();

  for (int kt = 0; kt < NTILE; ++kt) {
    const float* lb = lds_b[kt & 1];
    float* nxt = lds_b[(kt + 1) & 1];

#if VQ_ASYNC_LDS
    // Issue async staging of tile kt+1 (ASYNCcnt) before consuming tile kt.
    if (kt + 1 < NTILE) {
#pragma unroll
      for (int t = 0; t < 16; ++t) {
        int gi, li;
        vq_stage_addrs(t * 256 + tid, kt + 1, &gi, &li);
        __builtin_amdgcn_global_load_async_to_lds_b32(
            (int*)(emb + gi), (int*)&nxt[li], 0, 0);
      }
    }
#else
    float tmp[16];
    if (kt + 1 < NTILE) {
#pragma unroll
      for (int t = 0; t < 16; ++t) {
        int gi, li;
        vq_stage_addrs(t * 256 + tid, kt + 1, &gi, &li);
        tmp[t] = emb[gi];
      }
    }
#endif

    // ---- 64 chained f32 WMMAs, B register-pipelined in groups of 8 ----
    v8f acc = {};
    v2f bq[2][8];
#pragma unroll
    for (int u = 0; u < 8; ++u)
      bq[0][u] = *(const v2f*)&lb[u * 64 + boff];
#pragma unroll
    for (int g = 0; g < 8; ++g) {
      if (g < 7) {
#pragma unroll
        for (int u = 0; u < 8; ++u)
          bq[(g + 1) & 1][u] = *(const v2f*)&lb[((g + 1) * 8 + u) * 64 + boff];
      }
#pragma unroll
      for (int u = 0; u < 8; ++u) {
        acc = __builtin_amdgcn_wmma_f32_16x16x4_f32(
            /*neg_a=*/false, a[g * 8 + u], /*neg_b=*/false, bq[g & 1][u],
            /*c_mod=*/(short)0, acc, /*reuse_a=*/false, /*reuse_b=*/false);
      }
    }

    // argmin update: distance(m,k) = ||e_k||^2 - 2 x_m.e_k  (+ const per row)
    const int   kg = kt * 16 + l16;
    const float en = enorm[kg];
#pragma unroll
    for (int j = 0; j < 8; ++j) {
      float dist = en - 2.0f * acc[j];
      if (dist < best[j] || (dist == best[j] && kg < bidx[j])) {
        best[j] = dist;
        bidx[j] = kg;
      }
    }

#if VQ_ASYNC_LDS
    vq_wait_async0();
#else
    if (kt + 1 < NTILE) {
#pragma unroll
      for (int t = 0; t < 16; ++t) {
        int gi, li;
        vq_stage_addrs(t * 256 + tid, kt + 1, &gi, &li);
        nxt[li] = tmp[t];
      }
    }
#endif
    __syncthreads();
  }

  // Reduce across the 16 lanes holding the same row (xor masks stay in-half).
#pragma unroll
  for (int off = 8; off >= 1; off >>= 1) {
#pragma unroll
    for (int j = 0; j < 8; ++j) {
      float ov = __shfl_xor(best[j], off, 32);
      int   oi = __shfl_xor(bidx[j], off, 32);
      if (ov < best[j] || (ov == best[j] && oi < bidx[j])) {
        best[j] = ov;
        bidx[j] = oi;
      }
    }
  }
#pragma unroll
  for (int j = 0; j < 8; ++j) {
    if (l16 == 0) {
      int row = m_base + j + 8 * half;
      idx_out[row]   = bidx[j];
      idx_f_out[row] = (float)bidx[j];
    }
  }
}

// ---------------------------------------------------------------------------
// 3) gather quantized rows (coalesced via embT), loss partials, counts,
//    dw scatter into dwT[K,D] (block's 256 atomics hit one contiguous 1KB).
// ---------------------------------------------------------------------------
__global__ __launch_bounds__(256) void vq_gather_kernel(
    const float* __restrict__ x, const float* __restrict__ embT,
    const int* __restrict__ idx, float* __restrict__ q_out,
    float* __restrict__ dwT, float* __restrict__ counts,
    float* __restrict__ loss_acc) {
  const int n = blockIdx.x;
  const int d = threadIdx.x;
  const int k = idx[n];

  float q  = embT[(size_t)k * D_EMB + d];
  float xv = x[(size_t)n * D_EMB + d];
  q_out[(size_t)n * D_EMB + d] = q;

  float diff = q - xv;
  float sq = diff * diff;
#pragma unroll
  for (int off = 16; off >= 1; off >>= 1) sq += __shfl_xor(sq, off, 32);

  __shared__ float ls[8];
  if ((threadIdx.x & 31) == 0) ls[threadIdx.x >> 5] = sq;
  __syncthreads();
  if (threadIdx.x == 0) {
    float s = 0.f;
#pragma unroll
    for (int w = 0; w < 8; ++w) s += ls[w];
    atomicAdd(loss_acc, s);
    atomicAdd(&counts[k], 1.0f);
  }
  atomicAdd(&dwT[(size_t)k * D_EMB + d], xv);
}

// ---------------------------------------------------------------------------
// 4) scalar reductions: debias, n = sum(cs_avg), entropy -> loss, perplexity
// ---------------------------------------------------------------------------
__global__ __launch_bounds__(256) void vq_reduce_kernel(
    const float* __restrict__ counts, const float* __restrict__ ema_cs,
    const int* __restrict__ counter_p, float* __restrict__ scal,
    float* __restrict__ out_loss, float* __restrict__ out_perp) {
  const int tid = threadIdx.x;
  const float debias = 1.0f - powf(DECAY, (float)(counter_p[0] + 1));
  float s_n = 0.f, s_p = 0.f;
  for (int k = tid; k < K_EMB; k += 256) {
    float cnt = counts[k];
    s_n += (DECAY * ema_cs[k] + ONE_M_DECAY * cnt) / debias;
    float p = cnt / (float)N_ROWS;
    s_p += p * logf(p + 1e-10f);
  }
#pragma unroll
  for (int off = 16; off >= 1; off >>= 1) {
    s_n += __shfl_xor(s_n, off, 32);
    s_p += __shfl_xor(s_p, off, 32);
  }
  __shared__ float la[8], lb[8];
  if ((tid & 31) == 0) { la[tid >> 5] = s_n; lb[tid >> 5] = s_p; }
  __syncthreads();
  if (tid == 0) {
    float nt = 0.f, pl = 0.f;
#pragma unroll
    for (int w = 0; w < 8; ++w) { nt += la[w]; pl += lb[w]; }
    scal[1] = nt;
    scal[2] = debias;
    out_loss[0] = 0.25f * scal[0] / (float)((size_t)N_ROWS * D_EMB);
    out_perp[0] = expf(-pl);
  }
}

// ---------------------------------------------------------------------------
// 5) new_embeddings[d,k] = dw_avg / cs_norm   (coalesced over k)
// ---------------------------------------------------------------------------
__global__ __launch_bounds__(256) void vq_newemb_kernel(
    const float* __restrict__ emb, const float* __restrict__ ema_dw,
    const float* __restrict__ ema_cs, const float* __restrict__ counts,
    const float* __restrict__ dwT, const float* __restrict__ scal,
    const int* __restrict__ is_training_p, float* __restrict__ out) {
  size_t i = (size_t)blockIdx.x * 256 + threadIdx.x;  // i = d*K + k
  if (is_training_p[0] == 0) { out[i] = emb[i]; return; }
  int k = (int)(i & (K_EMB - 1));
  int d = (int)(i >> 13);  // K = 2^13
  float debias = scal[2];
  float n      = scal[1];
  float cs_avg  = (DECAY * ema_cs[k] + ONE_M_DECAY * counts[k]) / debias;
  float cs_norm = (cs_avg + EPS) / (n + (float)K_EMB * EPS) * n;
  float dw_avg  = (DECAY * ema_dw[i] +
                   ONE_M_DECAY * dwT[(size_t)k * D_EMB + d]) / debias;
  out[i] = dw_avg / cs_norm;
}

// ---------------------------------------------------------------------------
extern "C" void kernel_launch(void* const* d_in, const int* in_sizes, int n_in,
                              void* d_out, int out_size, void* d_ws,
                              size_t ws_size, hipStream_t stream) {
  (void)in_sizes; (void)n_in; (void)out_size; (void)ws_size;

  const float* x       = (const float*)d_in[0];  // inputs
  const float* emb     = (const float*)d_in[1];  // embeddings [D,K]
  const float* ema_cs  = (const float*)d_in[2];
  const float* ema_dw  = (const float*)d_in[3];
  const int*   counter = (const int*)d_in[4];
  const int*   is_trn  = (const int*)d_in[5];

  float* out = (float*)d_out;
  float* q_out    = out;                          // N*D
  float* loss_out = out + (size_t)N_ROWS * D_EMB; // 1
  float* perp_out = loss_out + 1;                 // 1
  float* idxf_out = perp_out + 1;                 // N
  float* nemb_out = idxf_out + N_ROWS;            // D*K

  float* ws     = (float*)d_ws;
  float* dwT    = ws + WS_DW;
  float* counts = ws + WS_CNT;
  float* scal   = ws + WS_SCAL;  // [0]=loss_acc [1]=n [2]=debias
  float* enorm  = ws + WS_ENORM;
  int*   idx_i  = (int*)(ws + WS_IDX);
  float* embT   = ws + WS_EMBT;

  (void)hipMemsetAsync(d_ws, 0, WS_ZERO_BYTES, stream);

  vq_enorm_kernel<<<K_EMB / 256, 256, 0, stream>>>(emb, enorm);

  vq_transpose_kernel<<<dim3(K_EMB / 32, D_EMB / 32), 256, 0, stream>>>(emb,
                                                                        embT);

  vq_argmin_kernel<<<N_ROWS / 128, 256, 0, stream>>>(x, emb, enorm, idx_i,
                                                     idxf_out);

  vq_gather_kernel<<<N_ROWS, 256, 0, stream>>>(x, embT, idx_i, q_out, dwT,
                                               counts, scal);

  vq_reduce_kernel<<<1, 256, 0, stream>>>(counts, ema_cs, counter, scal,
                                          loss_out, perp_out);

  vq_newemb_kernel<<<(D_EMB * K_EMB) / 256, 256, 0, stream>>>(
      emb, ema_dw, ema_cs, counts, dwT, scal, is_trn, nemb_out);
}